// Filters_33440615366820
// MI455X (gfx1250) — compile-verified
//
#include <hip/hip_runtime.h>

// CDNA5 / gfx1250, wave32. f32 WMMA (16x16x4) Toeplitz-conv + fused bias/maxpool.

typedef __attribute__((ext_vector_type(2))) float v2f;
typedef __attribute__((ext_vector_type(8))) float v8f;

#define NFILT  1024
#define HH     64
#define WW     64
#define KH     5
#define KW     5
#define LOCAL  3
#define WIN    11            // KH + 2*LOCAL
#define KPIX   121           // WIN*WIN
#define KPAD   124           // K rounded to multiple of 4 (31 wmma K-steps)
#define NPIX   36            // 6x6 conv outputs feeding the 2x2 maxpool
#define NPAD   48            // 3 N-tiles of 16
#define MB     64            // batches per workgroup (4 waves x 16)
#define THREADS 128

__global__ __launch_bounds__(THREADS)
void filters_wmma_kernel(const float* __restrict__ X,
                         const float* __restrict__ weight,
                         const float* __restrict__ bias,
                         const int*   __restrict__ positions,
                         float*       __restrict__ out)
{
    __shared__ float A_lds[MB * KPAD];       // 64 x 124 im2col windows (31 KB)
    __shared__ float B_lds[KPAD * NPAD];     // 124 x 48 Toeplitz weights (23.25 KB)

    const int f      = blockIdx.x;           // filter
    const int batch0 = blockIdx.y * MB;      // batch block base
    const int tid    = threadIdx.x;
    const int lane   = tid & 31;
    const int wave   = tid >> 5;
    const int half   = lane >> 4;            // 0: lanes 0-15, 1: lanes 16-31
    const int nl     = lane & 15;            // N-col within tile / M-row for A frag

    const int i0 = positions[2 * f + 0];     // window corner in padded image
    const int j0 = positions[2 * f + 1];

    // ---- Stage A: 11x11 window per batch, zero-padded (image OOB + K pad) ----
    for (int idx = tid; idx < MB * KPAD; idx += THREADS) {
        const int mb = idx / KPAD;
        const int k  = idx - mb * KPAD;
        float v = 0.0f;
        if (k < KPIX) {
            const int wi = k / WIN;
            const int wj = k - wi * WIN;
            const int r  = i0 + wi - LOCAL;  // unpadded image coords
            const int c  = j0 + wj - LOCAL;
            if ((unsigned)r < (unsigned)HH && (unsigned)c < (unsigned)WW)
                v = X[(size_t)(batch0 + mb) * (HH * WW) + r * WW + c];
        }
        A_lds[idx] = v;
    }

    // ---- Stage B: zero, then scatter the 5x5 weights into Toeplitz form ----
    for (int idx = tid; idx < KPAD * NPAD; idx += THREADS) B_lds[idx] = 0.0f;
    __syncthreads();
    for (int idx = tid; idx < NPIX * KH * KW; idx += THREADS) {
        const int p   = idx / (KH * KW);
        const int kk  = idx - p * (KH * KW);
        const int ki  = kk / KW;
        const int kj  = kk - ki * KW;
        const int pi  = p / 6;
        const int pj  = p - pi * 6;
        const int krow = (pi + ki) * WIN + (pj + kj);   // < 121, collision-free
        B_lds[krow * NPAD + p] = weight[f * (KH * KW) + kk];
    }
    __syncthreads();

    // ---- GEMM: D[16 batches x 48 pixels] = A[16 x 124] * Bt[124 x 48] ----
    // A frag (16x4 f32):  lane m=nl, elem r -> A[m][4s + 2*half + r]
    // B frag (4x16 f32):  lane n=nl, elem r -> B[4s + 2*half + r][16t + n]
    v8f acc0 = {}, acc1 = {}, acc2 = {};
    const float* Arow = &A_lds[(wave * 16 + nl) * KPAD];
    #pragma unroll 1
    for (int s = 0; s < KPAD / 4; ++s) {
        const int k0 = 4 * s + 2 * half;
        v2f a;  a.x  = Arow[k0];        a.y  = Arow[k0 + 1];
        const float* Br0 = &B_lds[k0 * NPAD + nl];
        const float* Br1 = Br0 + NPAD;
        v2f b0; b0.x = Br0[0];   b0.y = Br1[0];
        v2f b1; b1.x = Br0[16];  b1.y = Br1[16];
        v2f b2; b2.x = Br0[32];  b2.y = Br1[32];
        acc0 = __builtin_amdgcn_wmma_f32_16x16x4_f32(false, a, false, b0, (short)0, acc0, false, false);
        acc1 = __builtin_amdgcn_wmma_f32_16x16x4_f32(false, a, false, b1, (short)0, acc1, false, false);
        acc2 = __builtin_amdgcn_wmma_f32_16x16x4_f32(false, a, false, b2, (short)0, acc2, false, false);
    }

    // ---- Spill C tiles to LDS (reuse A region; all waves done reading) ----
    __syncthreads();
    float* conv = &A_lds[wave * (16 * NPAD)];   // 16 x 48 per wave (3072 fl total)
    #pragma unroll
    for (int r = 0; r < 8; ++r) {
        // C/D layout: elem r, lanes 0-15 -> M=r ; lanes 16-31 -> M=8+r ; N=16t+nl
        conv[(r + 8 * half) * NPAD +  0 + nl] = acc0[r];
        conv[(r + 8 * half) * NPAD + 16 + nl] = acc1[r];
        conv[(r + 8 * half) * NPAD + 32 + nl] = acc2[r];
    }
    __syncthreads();

    // ---- Bias + 3x3/3 maxpool -> [2,2], one float4 per (batch, filter) ----
    if (lane < 16) {
        const int b = batch0 + wave * 16 + nl;
        const float bv = bias[f];
        const float* cr = &conv[nl * NPAD];     // this batch's 36 conv values
        float4 o;
        float* oo = &o.x;
        #pragma unroll
        for (int q = 0; q < 4; ++q) {
            const int qi = q >> 1, qj = q & 1;
            float mx = -__builtin_inff();
            #pragma unroll
            for (int di = 0; di < 3; ++di)
                #pragma unroll
                for (int dj = 0; dj < 3; ++dj)
                    mx = fmaxf(mx, cr[(3 * qi + di) * 6 + (3 * qj + dj)]);
            oo[q] = mx + bv;
        }
        *(float4*)&out[(size_t)b * (NFILT * 4) + f * 4] = o;  // 16B-aligned
    }
}

extern "C" void kernel_launch(void* const* d_in, const int* in_sizes, int n_in,
                              void* d_out, int out_size, void* d_ws, size_t ws_size,
                              hipStream_t stream) {
    const float* X         = (const float*)d_in[0];   // [1024,1,64,64]
    const float* weight    = (const float*)d_in[1];   // [1024,1,5,5]
    const float* bias      = (const float*)d_in[2];   // [1024]
    const int*   positions = (const int*)d_in[3];     // [1024,2]
    float*       out       = (float*)d_out;           // [1024, 4096]

    dim3 grid(NFILT, 1024 / MB, 1);   // 1024 filters x 16 batch-blocks
    dim3 block(THREADS, 1, 1);
    filters_wmma_kernel<<<grid, block, 0, stream>>>(X, weight, bias, positions, out);
}